// CMCScore_infonce_11244224381542
// MI455X (gfx1250) — compile-verified
//
#include <hip/hip_runtime.h>
#include <math.h>

typedef __attribute__((ext_vector_type(2))) float v2f;
typedef __attribute__((ext_vector_type(8))) float v8f;

#define FEAT   128
#define NBANK  1000000
#define KP1    4097
#define BATCH  256
#define NTILE  257        /* ceil(4097/16) */
#define INV_T  2.0f       /* 1/T, T = 0.5 */

static const size_t OUT_P = (size_t)BATCH * KP1;   /* 1,048,832 */

/* workspace layout (floats): ln[0,32768) abn[32768,65536) sums[65536..65537] */

/* ---------------- kernel 1: L2 normalize l and ab ---------------- */
__global__ void norm_kernel(const float* __restrict__ l,
                            const float* __restrict__ ab,
                            float* __restrict__ ws)
{
    if (blockIdx.x == 0 && threadIdx.x == 0) { ws[65536] = 0.f; ws[65537] = 0.f; }
    int w    = (blockIdx.x * blockDim.x + threadIdx.x) >> 5;  /* wave id 0..511 */
    int lane = threadIdx.x & 31;
    if (w >= 2 * BATCH) return;
    const float* src = (w < BATCH) ? (l + (size_t)w * FEAT)
                                   : (ab + (size_t)(w - BATCH) * FEAT);
    float4 v = ((const float4*)src)[lane];
    float ss = v.x*v.x + v.y*v.y + v.z*v.z + v.w*v.w;
    #pragma unroll
    for (int off = 16; off; off >>= 1) ss += __shfl_xor(ss, off, 32);
    float inv = 1.0f / (sqrtf(ss) + 1e-7f);
    float4 o; o.x = v.x*inv; o.y = v.y*inv; o.z = v.z*inv; o.w = v.w*inv;
    ((float4*)(ws + (size_t)w * FEAT))[lane] = o;
}

/* ------------- kernel 2: gather rows, WMMA dot, exp, partial sum ------------- */
__global__ void score_kernel(const float* __restrict__ bank,  /* [NBANK,128]  */
                             const float* __restrict__ xn,    /* [256,128] normalized */
                             const int*   __restrict__ idx,   /* [256,4097]   */
                             float* __restrict__ P,           /* [256,4097]   */
                             float* __restrict__ sum_ptr)
{
    int wave = (blockIdx.x * blockDim.x + threadIdx.x) >> 5;
    int lane = threadIdx.x & 31;
    int half = lane >> 4;
    int m    = lane & 15;

    int b  = wave / NTILE;
    int kt = wave - b * NTILE;
    int k0 = kt * 16;

    int kk = k0 + m; if (kk > KP1 - 1) kk = KP1 - 1;   /* clamp tail */
    int row = idx[(size_t)b * KP1 + kk];
    const float* arow = bank + (size_t)row * FEAT;
    const float* brow = xn + (size_t)b * FEAT;

    /* A(16x4 f32): lane m, half h holds K = {4c+2h, 4c+2h+1} of gathered row m.
       B(4x16 f32): every column = ln[b] chunk -> all D columns equal the dot. */
    v8f c = {};
    #pragma unroll
    for (int ch = 0; ch < 32; ++ch) {
        int d = ch * 4 + half * 2;
        v2f a = *(const v2f*)(arow + d);
        v2f bb = *(const v2f*)(brow + d);
        c = __builtin_amdgcn_wmma_f32_16x16x4_f32(
                /*neg_a=*/false, a, /*neg_b=*/false, bb,
                /*c_mod=*/(short)0, c, /*reuse_a=*/false, /*reuse_b=*/false);
    }

    /* D layout: VGPR j, half h -> dot for tile row M = j + 8h (replicated over N) */
    float p[8];
    float s = 0.f;
    #pragma unroll
    for (int j = 0; j < 8; ++j) {
        int k = k0 + half * 8 + j;
        float e = __expf(c[j] * INV_T);
        p[j] = e;
        if (k < KP1) s += e;
    }

    if (m == 0) {   /* lanes 0 and 16 write the 16 scores */
        float* outp = P + (size_t)b * KP1 + k0 + half * 8;
        int nvalid = KP1 - (k0 + half * 8);
        if (nvalid > 8) nvalid = 8;
        for (int j = 0; j < nvalid; ++j) outp[j] = p[j];
    }

    /* wave sum = lane0 (m=0..7) + lane16 (m=8..15) contributions */
    float s0 = __shfl(s, 0, 32);
    float s1 = __shfl(s, 16, 32);
    if (lane == 0) atomicAdd(sum_ptr, s0 + s1);
}

/* ---------------- kernel 3: divide by Z = mean(P) * NBANK ---------------- */
__global__ void scale_kernel(float* __restrict__ out_l,
                             float* __restrict__ out_ab,
                             const float* __restrict__ sums)
{
    size_t i = (size_t)blockIdx.x * blockDim.x + threadIdx.x;
    if (i >= OUT_P) return;
    const float ratio = (float)((double)BATCH * KP1 / (double)NBANK);
    float invl  = ratio / sums[0];
    float invab = ratio / sums[1];
    out_l[i]  *= invl;
    out_ab[i] *= invab;
}

/* ---------------- kernel 4: copy both memory banks to output ---------------- */
__global__ void copy_kernel(const float4* __restrict__ srcA,
                            const float4* __restrict__ srcB,
                            float4* __restrict__ dstA,
                            float4* __restrict__ dstB)
{
    const size_t n = (size_t)NBANK * (FEAT / 4);   /* 32M float4 per bank */
    size_t stride = (size_t)gridDim.x * blockDim.x;
    for (size_t i = (size_t)blockIdx.x * blockDim.x + threadIdx.x; i < n; i += stride) {
        dstA[i] = srcA[i];
        dstB[i] = srcB[i];
    }
}

/* ------- kernel 5: momentum-update the scattered rows (after the copy) ------- */
__global__ void update_kernel(const float* __restrict__ mem_l,
                              const float* __restrict__ mem_ab,
                              const float* __restrict__ ws,   /* ln then abn */
                              const int*   __restrict__ y,
                              float* __restrict__ out_mem_l,
                              float* __restrict__ out_mem_ab)
{
    int w    = (blockIdx.x * blockDim.x + threadIdx.x) >> 5;  /* 0..511 */
    int lane = threadIdx.x & 31;
    if (w >= 2 * BATCH) return;
    int bank = w >> 8;          /* 0 = l-bank, 1 = ab-bank */
    int b    = w & 255;
    int yb   = y[b];
    /* determinism with duplicate y: only the last occurrence writes */
    for (int b2 = b + 1; b2 < BATCH; ++b2)
        if (y[b2] == yb) return;

    const float* mem = bank ? mem_ab : mem_l;
    const float* xn  = ws + ((size_t)bank * BATCH + b) * FEAT;
    float* outm = (bank ? out_mem_ab : out_mem_l) + (size_t)yb * FEAT;

    float4 mv = ((const float4*)(mem + (size_t)yb * FEAT))[lane];
    float4 xv = ((const float4*)xn)[lane];
    float4 p;
    p.x = mv.x * 0.5f + xv.x * 0.5f;
    p.y = mv.y * 0.5f + xv.y * 0.5f;
    p.z = mv.z * 0.5f + xv.z * 0.5f;
    p.w = mv.w * 0.5f + xv.w * 0.5f;
    float ss = p.x*p.x + p.y*p.y + p.z*p.z + p.w*p.w;
    #pragma unroll
    for (int off = 16; off; off >>= 1) ss += __shfl_xor(ss, off, 32);
    float inv = 1.0f / sqrtf(ss);    /* no eps here, matches reference */
    float4 o; o.x = p.x*inv; o.y = p.y*inv; o.z = p.z*inv; o.w = p.w*inv;
    ((float4*)outm)[lane] = o;
}

extern "C" void kernel_launch(void* const* d_in, const int* in_sizes, int n_in,
                              void* d_out, int out_size, void* d_ws, size_t ws_size,
                              hipStream_t stream)
{
    (void)in_sizes; (void)n_in; (void)out_size; (void)ws_size;

    const float* l      = (const float*)d_in[0];
    const float* ab     = (const float*)d_in[1];
    const float* mem_l  = (const float*)d_in[2];
    const float* mem_ab = (const float*)d_in[3];
    const int*   y      = (const int*)d_in[4];
    const int*   idx    = (const int*)d_in[5];

    float* out        = (float*)d_out;
    float* out_l      = out;
    float* out_ab     = out + OUT_P;
    float* out_mem_l  = out + 2 * OUT_P;
    float* out_mem_ab = out + 2 * OUT_P + (size_t)NBANK * FEAT;

    float* ws   = (float*)d_ws;
    float* sums = ws + 65536;

    /* 1) normalize (512 waves) + zero accumulators */
    norm_kernel<<<128, 128, 0, stream>>>(l, ab, ws);

    /* 2) scores: 256*257 = 65792 waves, 4 waves/block -> 16448 blocks exactly */
    int blocks = (BATCH * NTILE) / 4;
    score_kernel<<<blocks, 128, 0, stream>>>(mem_ab, ws,         idx, out_l,  sums + 0);
    score_kernel<<<blocks, 128, 0, stream>>>(mem_l,  ws + 32768, idx, out_ab, sums + 1);

    /* 3) normalize by Z (4097*256 == OUT_P exactly) */
    scale_kernel<<<4097, 256, 0, stream>>>(out_l, out_ab, sums);

    /* 4) bulk bank copy (memory-bound; float4 grid-stride) */
    copy_kernel<<<2048, 256, 0, stream>>>((const float4*)mem_l, (const float4*)mem_ab,
                                          (float4*)out_mem_l, (float4*)out_mem_ab);

    /* 5) scatter momentum rows over the copies (512 waves) */
    update_kernel<<<128, 128, 0, stream>>>(mem_l, mem_ab, ws, y,
                                           out_mem_l, out_mem_ab);
}